// Operator_46050639347872
// MI455X (gfx1250) — compile-verified
//
#include <hip/hip_runtime.h>
#include <hip/hip_bf16.h>
#include <math.h>

typedef float v2f __attribute__((ext_vector_type(2)));
typedef float v8f __attribute__((ext_vector_type(8)));

#define KVT_ELEMS (512 * 512)   // per (B,T') tile matrix

__device__ __forceinline__ float sqrt_act(float v) {
    return copysignf(sqrtf(fabsf(v)), v);
}

// ---------------------------------------------------------------------------
// K0: q[B,C,X] = sum_{x,c} Q[X,C,x,c] * x[B,0,11,c,x]   (4096 outputs)
//     also zero tmp[800] for the atomic accumulation in k_tmp.
// ---------------------------------------------------------------------------
__global__ void k_q(const float* __restrict__ xin, const float* __restrict__ Qm,
                    float* __restrict__ q, float* __restrict__ tmp) {
    int gid = blockIdx.x * blockDim.x + threadIdx.x;   // 0..4095
    if (gid < 800) tmp[gid] = 0.f;
    int B = gid >> 9;
    int C = (gid >> 6) & 7;
    int X = gid & 63;
    const float* Qrow = Qm + (size_t)(X * 8 + C) * 512;
    const float* xv = xin + ((size_t)B * 12 + 11) * 512;   // last T slice
    float acc = 0.f;
    for (int t = 0; t < 512; ++t) {
        int c = t & 7, xx = t >> 3;
        acc += Qrow[t] * xv[c * 64 + xx];
    }
    q[gid] = acc;
}

// ---------------------------------------------------------------------------
// K1: kvt[bt][r][s] = sqrt_act( sum_t K[r][t]*w[bt][t]*V[s][t] )
//     16 independent 512x512x512 fp32 GEMMs via V_WMMA_F32_16X16X4_F32.
//     One wave per 16x16 output tile; 8 waves per block share w in LDS.
// ---------------------------------------------------------------------------
__global__ void k_kvt(const float* __restrict__ xin, const float* __restrict__ Km,
                      const float* __restrict__ Vm, float* __restrict__ kvt) {
    __shared__ float wlds[512];
    int bt = blockIdx.x >> 7;          // 0..15 : (B,T')
    int group = blockIdx.x & 127;      // 128 groups of 8 tiles
    int B = bt >> 1, Tp = bt & 1;
    int xbase = ((B * 12) + 10 + Tp) * 512;     // x2 = x[:,:,10+Tp]
    for (int i = threadIdx.x; i < 512; i += 256) {
        // w[t] with t = x*8 + c  ->  x2[c,x]
        wlds[i] = xin[xbase + (i & 7) * 64 + (i >> 3)];
    }
    __syncthreads();

    int wave = threadIdx.x >> 5;
    int lane = threadIdx.x & 31;
    int tile = group * 8 + wave;       // 0..1023
    int r0 = (tile >> 5) << 4;
    int s0 = (tile & 31) << 4;
    int m = lane & 15;                 // A-row / B-col / D-col for this lane
    int hi = lane >> 4;
    const float* Arow = Km + (size_t)(r0 + m) * 512;
    const float* Brow = Vm + (size_t)(s0 + m) * 512;
    int kA = hi * 2;                   // A: VGPR0 holds K = {0,1} | {2,3}
    int kB = hi;                       // B: rows K striped across lane halves

    v8f acc = {};
    for (int k0 = 0; k0 < 512; k0 += 4) {
        v2f a, b;
        a.x = Arow[k0 + kA]     * wlds[k0 + kA];
        a.y = Arow[k0 + kA + 1] * wlds[k0 + kA + 1];
        b.x = Brow[k0 + kB];
        b.y = Brow[k0 + kB + 2];
        acc = __builtin_amdgcn_wmma_f32_16x16x4_f32(false, a, false, b,
                                                    (short)0, acc, false, false);
    }

    float* outp = kvt + (size_t)bt * KVT_ELEMS;
    int mM = hi * 8;                   // D: VGPR j -> M = j + 8*hi, N = lane&15
    for (int j = 0; j < 8; ++j) {
        outp[(size_t)(r0 + mM + j) * 512 + (s0 + m)] = sqrt_act(acc[j]);
    }
}

// ---------------------------------------------------------------------------
// K2: tmp[B,n] = sum_{C,K,X,Y} Ano[C,K,X,Y,n] * kvt_n[B,C,K,X,Y]
//     Block = 1024-element chunk of i=(C,K,X,Y); kvt chunk (8 batches) staged
//     in LDS with index permutation; Ano streamed exactly once, coalesced in n.
// ---------------------------------------------------------------------------
#define CHUNK 1024
__global__ void k_tmp(const float* __restrict__ Ano, const float* __restrict__ kvt,
                      float* __restrict__ tmp) {
    __shared__ float kch[8 * CHUNK];   // 32 KB
    int cb = blockIdx.x;               // 0..255
    for (int idx = threadIdx.x; idx < 8 * CHUNK; idx += 256) {
        int b = idx >> 10;
        int k = idx & (CHUNK - 1);
        int i = cb * CHUNK + k;        // i = ((C*8+K)*64+X)*64+Y
        int C = i >> 15, Kk = (i >> 12) & 7, X = (i >> 6) & 63, Y = i & 63;
        kch[idx] = kvt[(size_t)(b * 2) * KVT_ELEMS
                       + (size_t)((X << 3) + C) * 512 + ((Y << 3) + Kk)];
    }
    __syncthreads();
    int n = threadIdx.x;
    if (n < 100) {
        float acc[8] = {};
        const float* ap = Ano + (size_t)cb * CHUNK * 100 + n;
        for (int k = 0; k < CHUNK; ++k) {
            float a = ap[(size_t)k * 100];
            #pragma unroll
            for (int b = 0; b < 8; ++b) acc[b] += a * kch[b * CHUNK + k];
        }
        #pragma unroll
        for (int b = 0; b < 8; ++b) unsafeAtomicAdd(&tmp[b * 100 + n], acc[b]);
    }
}

// ---------------------------------------------------------------------------
// K3: S[C,c,X,n] = sum_x Aon[C,c,X,x,n]          (409600 outputs)
// ---------------------------------------------------------------------------
__global__ void k_S(const float* __restrict__ Aon, float* __restrict__ S) {
    int o = blockIdx.x * 256 + threadIdx.x;        // < 409600
    int n = o % 100;
    int rest = o / 100;                            // (C*8+c)*64+X
    const float* ap = Aon + (size_t)rest * 6400 + n;
    float acc = 0.f;
    for (int x = 0; x < 64; ++x) acc += ap[x * 100];
    S[o] = acc;
}

// ---------------------------------------------------------------------------
// K4: G[B,c,n] = sum_{C,X} S[C,c,X,n] * q[B,C,X]  (6400 outputs)
// ---------------------------------------------------------------------------
__global__ void k_G(const float* __restrict__ S, const float* __restrict__ q,
                    float* __restrict__ G) {
    int o = blockIdx.x * 256 + threadIdx.x;        // < 6400
    int n = o % 100;
    int bc = o / 100;                              // B*8+c
    int B = bc >> 3, c = bc & 7;
    float acc = 0.f;
    for (int C = 0; C < 8; ++C)
        for (int X = 0; X < 64; ++X)
            acc += S[(size_t)((C * 8 + c) * 64 + X) * 100 + n]
                 * q[(B * 8 + C) * 64 + X];
    G[o] = acc;
}

// ---------------------------------------------------------------------------
// K5: y[B,c] = sum_{C,X,Y} q[B,C,X]*Aoo[C,c,X,Y]*kvt_o[B,C,c,X,Y]
//            + sum_n tmp[B,n]*G[B,c,n]
// ---------------------------------------------------------------------------
__global__ void k_y(const float* __restrict__ q, const float* __restrict__ Aoo,
                    const float* __restrict__ kvt, const float* __restrict__ tmp,
                    const float* __restrict__ G, float* __restrict__ y) {
    __shared__ float red[256];
    int B = blockIdx.x >> 3, c = blockIdx.x & 7;
    const float* kvo = kvt + (size_t)(B * 2 + 1) * KVT_ELEMS;   // T'=1 slice
    float acc = 0.f;
    for (int j = threadIdx.x; j < 32768; j += 256) {
        int C = j >> 12, X = (j >> 6) & 63, xx = j & 63;
        float aoo = Aoo[(size_t)((C * 8 + c) * 64 + X) * 64 + xx];
        float kv  = kvo[(size_t)((X << 3) + C) * 512 + ((xx << 3) + c)];
        acc += q[(B * 8 + C) * 64 + X] * aoo * kv;
    }
    if (threadIdx.x < 100)
        acc += tmp[B * 100 + threadIdx.x] * G[(size_t)blockIdx.x * 100 + threadIdx.x];
    red[threadIdx.x] = acc;
    __syncthreads();
    for (int s = 128; s > 0; s >>= 1) {
        if (threadIdx.x < s) red[threadIdx.x] += red[threadIdx.x + s];
        __syncthreads();
    }
    if (threadIdx.x == 0) y[blockIdx.x] = red[0];
}

// ---------------------------------------------------------------------------
extern "C" void kernel_launch(void* const* d_in, const int* in_sizes, int n_in,
                              void* d_out, int out_size, void* d_ws, size_t ws_size,
                              hipStream_t stream) {
    const float* x   = (const float*)d_in[0];
    const float* Km  = (const float*)d_in[1];
    const float* Qm  = (const float*)d_in[2];
    const float* Vm  = (const float*)d_in[3];
    const float* Aoo = (const float*)d_in[4];
    const float* Aon = (const float*)d_in[5];
    const float* Ano = (const float*)d_in[6];

    float* ws  = (float*)d_ws;
    float* kvt = ws;                              // 16 * 262144 floats
    float* q   = kvt + (size_t)16 * KVT_ELEMS;    // 4096
    float* tmp = q + 4096;                        // 800
    float* S   = tmp + 800;                       // 409600
    float* G   = S + 409600;                      // 6400
    float* y   = (float*)d_out;                   // 64 floats

    k_q  <<<16,   256, 0, stream>>>(x, Qm, q, tmp);
    k_kvt<<<2048, 256, 0, stream>>>(x, Km, Vm, kvt);
    k_S  <<<1600, 256, 0, stream>>>(Aon, S);
    k_tmp<<<256,  256, 0, stream>>>(Ano, kvt, tmp);
    k_G  <<<25,   256, 0, stream>>>(S, q, G);
    k_y  <<<64,   256, 0, stream>>>(q, Aoo, kvt, tmp, G, y);
}